// ConvexFusionPrimalDual_75935021793859
// MI455X (gfx1250) — compile-verified
//
#include <hip/hip_runtime.h>
#include <hip/hip_bf16.h>
#include <stdint.h>

#define BN 4
#define KN 8
#define HN 256
#define WN 256
#define TILE 16
#define HW ((size_t)HN * WN)
#define NEL ((size_t)BN * KN * HW)

#define TAU_C 0.25f
#define SIG_C 0.25f
#define BETA_C 0.2f
#define CC 0.5f          // c = tau * (lam_c + lam_d) = 0.25 * 2
#define RCC 2.0f         // 1/c

#if __has_builtin(__builtin_amdgcn_global_load_async_to_lds_b32)
#define USE_ASYNC_LDS 1
typedef __attribute__((address_space(1))) int as1_int;
typedef __attribute__((address_space(3))) int as3_int;
#endif

__device__ __forceinline__ float frcp(float x) { return __builtin_amdgcn_rcpf(x); }

// ---------------------------------------------------------------- init ------
__global__ __launch_bounds__(256) void pd_init(
    const float* __restrict__ pC, const float* __restrict__ pD,
    float* __restrict__ blog, float* __restrict__ q0, float* __restrict__ qb0,
    float* __restrict__ px0, float* __restrict__ py0)
{
    size_t p = (size_t)blockIdx.x * 256 + threadIdx.x;        // pixel id < B*H*W
    size_t bb = p >> 16, pix = p & 65535;
    size_t base = bb * (KN * HW) + pix;

    float c[KN], d[KN], bl[KN];
    float sc = 0.f, sd = 0.f;
#pragma unroll
    for (int k = 0; k < KN; ++k) {
        c[k] = fmaxf(pC[base + (size_t)k * HW], 1e-12f); sc += c[k];
        d[k] = fmaxf(pD[base + (size_t)k * HW], 1e-12f); sd += d[k];
    }
    float rc = frcp(sc), rd = frcp(sd);
    float bmax = -1e30f;
#pragma unroll
    for (int k = 0; k < KN; ++k) {
        float bk = __logf(c[k] * rc) + __logf(d[k] * rd);
        bl[k] = bk;
        bmax = fmaxf(bmax, 0.5f * bk);
    }
    float e[KN], s = 0.f;
#pragma unroll
    for (int k = 0; k < KN; ++k) { e[k] = __expf(0.5f * bl[k] - bmax); s += e[k]; }
    float rs = frcp(s);
#pragma unroll
    for (int k = 0; k < KN; ++k) {
        size_t o = base + (size_t)k * HW;
        float qq = e[k] * rs;
        blog[o] = bl[k]; q0[o] = qq; qb0[o] = qq; px0[o] = 0.f; py0[o] = 0.f;
    }
}

// ------------------------------------------------------------- iterate ------
__global__ __launch_bounds__(256) void pd_iter(
    const float* __restrict__ q_src, const float* __restrict__ qb_src,
    const float* __restrict__ px_src, const float* __restrict__ py_src,
    float* __restrict__ q_dst, float* __restrict__ qb_dst,
    float* __restrict__ px_dst, float* __restrict__ py_dst,
    const float* __restrict__ blog,
    const float* __restrict__ wx, const float* __restrict__ wy)
{
    __shared__ float sQ[KN * 18 * 18];                 // q_bar tile + 1-halo
    const int bb = blockIdx.z;
    const int tx0 = blockIdx.x * TILE, ty0 = blockIdx.y * TILE;
    const int lx = threadIdx.x, ly = threadIdx.y;
    const int tid = ly * TILE + lx;
    const float* qbB = qb_src + (size_t)bb * (KN * HW);

    // ---- stage q_bar halo tile into LDS (CDNA5 async global->LDS path) ----
    for (int e = tid; e < KN * 18 * 18; e += 256) {
        int ch = e / 324, rem = e - ch * 324;
        int r = rem / 18, cx = rem - r * 18;
        int gy = ty0 + r - 1;  gy = gy < 0 ? 0 : (gy >= HN ? HN - 1 : gy);
        int gx = tx0 + cx - 1; gx = gx < 0 ? 0 : (gx >= WN ? WN - 1 : gx);
        const float* gp = qbB + (size_t)ch * HW + (size_t)gy * WN + gx;
#ifdef USE_ASYNC_LDS
        __builtin_amdgcn_global_load_async_to_lds_b32(
            (as1_int*)(uintptr_t)gp,
            (as3_int*)(uintptr_t)(&sQ[e]),
            0, 0);
#else
        sQ[e] = *gp;
#endif
    }
#ifdef USE_ASYNC_LDS
#if __has_builtin(__builtin_amdgcn_s_wait_asynccnt)
    __builtin_amdgcn_s_wait_asynccnt(0);
#else
    asm volatile("s_wait_asynccnt 0" ::: "memory");
#endif
#endif
    __syncthreads();

    const int x = tx0 + lx, y = ty0 + ly;
    const size_t pix = (size_t)y * WN + x;
    const size_t bK = (size_t)bb * (KN * HW);

    const float limx_c = (x < WN - 1) ? BETA_C * wx[((size_t)bb * HN + y) * (WN - 1) + x] : 0.f;
    const float limx_m = (x > 0)      ? BETA_C * wx[((size_t)bb * HN + y) * (WN - 1) + (x - 1)] : 0.f;
    const float limy_c = (y < HN - 1) ? BETA_C * wy[((size_t)bb * (HN - 1) + y) * WN + x] : 0.f;
    const float limy_m = (y > 0)      ? BETA_C * wy[((size_t)bb * (HN - 1) + (y - 1)) * WN + x] : 0.f;

    float d0[KN], qold[KN];
    float dmin = 1e30f, dmax = -1e30f;
#pragma unroll
    for (int k = 0; k < KN; ++k) {
        const size_t base = bK + (size_t)k * HW + pix;
        const int sb = k * 324;
        float qbC = sQ[sb + (ly + 1) * 18 + (lx + 1)];
        float qbL = sQ[sb + (ly + 1) * 18 + lx];
        float qbR = sQ[sb + (ly + 1) * 18 + (lx + 2)];
        float qbU = sQ[sb + ly * 18 + (lx + 1)];
        float qbD = sQ[sb + (ly + 2) * 18 + (lx + 1)];

        float px_c = 0.f, px_m = 0.f, py_c = 0.f, py_m = 0.f;
        if (x < WN - 1) { float t = px_src[base] + SIG_C * (qbR - qbC);
                          px_c = fminf(fmaxf(t, -limx_c), limx_c); }
        if (x > 0)      { float t = px_src[base - 1] + SIG_C * (qbC - qbL);
                          px_m = fminf(fmaxf(t, -limx_m), limx_m); }
        if (y < HN - 1) { float t = py_src[base] + SIG_C * (qbD - qbC);
                          py_c = fminf(fmaxf(t, -limy_c), limy_c); }
        if (y > 0)      { float t = py_src[base - WN] + SIG_C * (qbC - qbU);
                          py_m = fminf(fmaxf(t, -limy_m), limy_m); }
        px_dst[base] = px_c;                 // x == W-1 column stays 0 (padded)
        py_dst[base] = py_c;                 // y == H-1 row stays 0 (padded)

        float div = (px_m - px_c) + (py_m - py_c);   // adjoint divergence
        float qo = q_src[base];
        float v = qo - TAU_C * div;
        float dk = v + TAU_C * blog[base] - CC;      // d0 = v + tau*b - c
        d0[k] = dk; qold[k] = qo;
        dmin = fminf(dmin, dk); dmax = fmaxf(dmax, dk);
    }

    // ---- bisection on mu; q_of_mu uses 5-step Lambert-W Newton ----
    float mu_l = dmin - 10.0f, mu_h = dmax + 10.0f;  // +/- 20*c
    float qf[KN];
    for (int it = 0; it <= 10; ++it) {
        float mu = 0.5f * (mu_l + mu_h);
        float s = 0.f;
#pragma unroll
        for (int k = 0; k < KN; ++k) {
            float z  = fminf(fmaxf((d0[k] - mu) * RCC, -50.f), 50.f);
            float zz = __expf(z) * RCC;              // exp(z)/c
            float w  = __logf(zz + 1.0f);
#pragma unroll
            for (int j = 0; j < 5; ++j) {
                float ew = __expf(w);
                w -= (w * ew - zz) * frcp(ew * (w + 1.0f) + 1e-12f);
            }
            float qv = fmaxf(CC * w, 0.f);
            qf[k] = qv; s += qv;
        }
        if (it < 10) {
            bool gt = s > 1.0f;
            mu_l = gt ? mu : mu_l;
            mu_h = gt ? mu_h : mu;
        } else {
            float inv = frcp(fmaxf(s, 1e-12f));
#pragma unroll
            for (int k = 0; k < KN; ++k) {
                const size_t base = bK + (size_t)k * HW + pix;
                float qn = qf[k] * inv;
                q_dst[base]  = qn;
                qb_dst[base] = 2.0f * qn - qold[k];  // theta = 1 over-relaxation
            }
        }
    }
}

// ------------------------------------------------------------- finalize ----
__global__ __launch_bounds__(256) void pd_final(
    const float* __restrict__ q, float* __restrict__ out)
{
    size_t p = (size_t)blockIdx.x * 256 + threadIdx.x;       // pixel id
    size_t bb = p >> 16, pix = p & 65535;
    size_t base = bb * (KN * HW) + pix;
    float best = -1e30f; int bi = 0;
#pragma unroll
    for (int k = 0; k < KN; ++k) {
        float v = q[base + (size_t)k * HW];
        out[base + (size_t)k * HW] = v;
        if (v > best) { best = v; bi = k; }
    }
    out[NEL + p] = (float)bi;                                 // argmax plane
}

// ------------------------------------------------------------- launcher ----
extern "C" void kernel_launch(void* const* d_in, const int* in_sizes, int n_in,
                              void* d_out, int out_size, void* d_ws, size_t ws_size,
                              hipStream_t stream) {
    const float* pC = (const float*)d_in[0];
    const float* pD = (const float*)d_in[1];
    const float* wx = (const float*)d_in[2];
    const float* wy = (const float*)d_in[3];
    float* out = (float*)d_out;
    float* ws = (float*)d_ws;

    float* blog   = ws;
    float* q_[2]  = { ws + 1 * NEL, ws + 2 * NEL };
    float* qb_[2] = { ws + 3 * NEL, ws + 4 * NEL };
    float* px_[2] = { ws + 5 * NEL, ws + 6 * NEL };
    float* py_[2] = { ws + 7 * NEL, ws + 8 * NEL };

    const int pixBlocks = (int)((size_t)BN * HW / 256);       // 1024
    pd_init<<<pixBlocks, 256, 0, stream>>>(pC, pD, blog, q_[0], qb_[0], px_[0], py_[0]);

    dim3 gb(WN / TILE, HN / TILE, BN), tb(TILE, TILE);
    int cur = 0;
    for (int it = 0; it < 80; ++it) {
        int nxt = cur ^ 1;
        pd_iter<<<gb, tb, 0, stream>>>(q_[cur], qb_[cur], px_[cur], py_[cur],
                                       q_[nxt], qb_[nxt], px_[nxt], py_[nxt],
                                       blog, wx, wy);
        cur = nxt;
    }
    pd_final<<<pixBlocks, 256, 0, stream>>>(q_[cur], out);
}